// MambaBlock_55937654063154
// MI455X (gfx1250) — compile-verified
//
#include <hip/hip_runtime.h>
#include <math.h>

typedef __attribute__((ext_vector_type(2))) float v2f;
typedef __attribute__((ext_vector_type(4))) float v4f;
typedef __attribute__((ext_vector_type(8))) float v8f;

#define D_MODEL 1024
#define D_STATE 16
#define D_CONV  4
#define D_INNER 2048
#define DT_RANK 64
#define BATCH   2
#define SEQ     2048
#define MROWS   (BATCH * SEQ)   // 4096 rows for every GEMM (multiple of 64)

__device__ __forceinline__ float silu_f(float x) {
    return x / (1.0f + expf(-x));
}
__device__ __forceinline__ float softplus_f(float x) {
    return (x > 20.0f) ? x : log1pf(expf(x));
}

struct Frag { v2f a0, a1, b0, b1; };

// ---------------------------------------------------------------------------
// C[M,N] = epilogue( A[M,K] (row-major, lda) * B[N,K]^T (row-major, ldb) )
//
// fp32 WMMA (V_WMMA_F32_16X16X4_F32), wave32. Each wave computes a 32x32
// output patch = 2x2 grid of 16x16 tiles in four v8f accumulators.
// Per K-step of 4: 2 A-frag + 2 B-frag b64 loads feed 4 WMMAs (1 load/WMMA).
// A 3-stage software pipeline keeps ~8-12 loads in flight at every WMMA
// (explicit buffer rotation; #pragma unroll 6 renames the rotation away).
// The loop is branch-free: EXEC stays all-1s around every WMMA.
//
// Fragment layout (documented A 16x4 / B 4x16 / C 16x16 f32 layouts):
//   A frag: lane<16 -> {A[m0+lr][k0], A[m0+lr][k0+1]}, lane>=16 -> k0+2,k0+3
//   B frag: same pattern on W[n0+lr][...] since B[k][n] = W[n][k]
//   C tile: element (m0 + v + 8*half, n0 + lr) = acc[v]
//
// Requires: M % 64 == 0, N % 32 == 0, K % 4 == 0, K >= 12 (all call sites:
// N = 4096 / 96 / 2048 / 1024; K = 1024 / 2048 / 64). Out-of-range waves
// return whole (wave-uniform guard).
// mode 0: plain store; mode 1: softplus(acc + bias[n]).
// ---------------------------------------------------------------------------
__global__ void wmma_gemm_nt(const float* __restrict__ A, const float* __restrict__ B,
                             float* __restrict__ C, const float* __restrict__ bias,
                             int M, int N, int K, int lda, int ldb, int ldc, int mode)
{
    const int lane = threadIdx.x & 31;
    const int wave = threadIdx.x >> 5;     // 4 waves -> 2x2 grid of 32x32 patches
    const int half = lane >> 4;
    const int lr   = lane & 15;

    const int m0 = (blockIdx.y * 2 + (wave >> 1)) * 32;
    const int n0 = (blockIdx.x * 2 + (wave & 1)) * 32;
    if (n0 + 32 > N) return;               // whole-wave guard (N % 32 == 0)

    const float* arow0 = A + (size_t)(m0 + lr) * lda + 2 * half;
    const float* arow1 = arow0 + (size_t)16 * lda;
    const float* brow0 = B + (size_t)(n0 + lr) * ldb + 2 * half;
    const float* brow1 = brow0 + (size_t)16 * ldb;

#define LDFRAG(f, kk)                              \
    (f).a0 = *(const v2f*)(arow0 + (kk));          \
    (f).a1 = *(const v2f*)(arow1 + (kk));          \
    (f).b0 = *(const v2f*)(brow0 + (kk));          \
    (f).b1 = *(const v2f*)(brow1 + (kk));

#define DOWMMA(f)                                                              \
    acc00 = __builtin_amdgcn_wmma_f32_16x16x4_f32(false, (f).a0, false, (f).b0,\
                                                  (short)0, acc00, false, false);\
    acc10 = __builtin_amdgcn_wmma_f32_16x16x4_f32(false, (f).a1, false, (f).b0,\
                                                  (short)0, acc10, false, false);\
    acc01 = __builtin_amdgcn_wmma_f32_16x16x4_f32(false, (f).a0, false, (f).b1,\
                                                  (short)0, acc01, false, false);\
    acc11 = __builtin_amdgcn_wmma_f32_16x16x4_f32(false, (f).a1, false, (f).b1,\
                                                  (short)0, acc11, false, false);

    v8f acc00 = {0.f,0.f,0.f,0.f,0.f,0.f,0.f,0.f};
    v8f acc01 = acc00, acc10 = acc00, acc11 = acc00;

    Frag f0, f1, f2;
    LDFRAG(f0, 0);
    LDFRAG(f1, 4);
    LDFRAG(f2, 8);

    int k = 0;
    #pragma unroll 6
    for (; k + 12 < K; k += 4) {
        Frag fn;
        LDFRAG(fn, k + 12);       // issue next loads before draining oldest
        DOWMMA(f0);
        f0 = f1; f1 = f2; f2 = fn;
    }
    DOWMMA(f0);
    DOWMMA(f1);
    DOWMMA(f2);

#undef LDFRAG
#undef DOWMMA

    const float bv0 = (mode == 1 && bias) ? bias[n0 + lr] : 0.0f;
    const float bv1 = (mode == 1 && bias) ? bias[n0 + 16 + lr] : 0.0f;
    #pragma unroll
    for (int v = 0; v < 8; ++v) {
        const int mA = m0 + v + 8 * half;        // first M-subtile rows
        const int mB = mA + 16;                  // second M-subtile rows
        float x00 = acc00[v], x10 = acc10[v];
        float x01 = acc01[v], x11 = acc11[v];
        if (mode == 1) {
            x00 = softplus_f(x00 + bv0); x10 = softplus_f(x10 + bv0);
            x01 = softplus_f(x01 + bv1); x11 = softplus_f(x11 + bv1);
        }
        C[(size_t)mA * ldc + (n0 + lr)]      = x00;
        C[(size_t)mB * ldc + (n0 + lr)]      = x10;
        C[(size_t)mA * ldc + (n0 + 16 + lr)] = x01;
        C[(size_t)mB * ldc + (n0 + 16 + lr)] = x11;
    }
}

// ---------------------------------------------------------------------------
// Depthwise causal conv1d (K=4) over the x_in half of xz, + bias, SiLU.
// xz layout: (MROWS, 2*D_INNER); x_in = columns [0, D_INNER).
// ---------------------------------------------------------------------------
__global__ void conv_silu_kernel(const float* __restrict__ xz, const float* __restrict__ cw,
                                 const float* __restrict__ cb, float* __restrict__ xssm)
{
    const int i = blockIdx.x * blockDim.x + threadIdx.x;
    if (i >= MROWS * D_INNER) return;
    const int d   = i & (D_INNER - 1);
    const int row = i >> 11;               // /D_INNER
    const int l   = row & (SEQ - 1);

    float acc = cb[d];
    #pragma unroll
    for (int k = 0; k < D_CONV; ++k) {
        const int ll = l - (D_CONV - 1) + k;           // causal left pad
        if (ll >= 0)
            acc += cw[d * D_CONV + k] *
                   xz[(size_t)(row - (D_CONV - 1) + k) * (2 * D_INNER) + d];
    }
    xssm[i] = silu_f(acc);
}

// ---------------------------------------------------------------------------
// Selective scan. One thread per (batch, channel d); h[0..15] in registers.
// proj layout per row (96 floats): dt[0..63] | Bm[64..79] | Cm[80..95].
// Bm/Cm rows are 16B-aligned -> four float4 loads each, broadcast via cache.
// ---------------------------------------------------------------------------
__global__ void scan_kernel(const float* __restrict__ delta, const float* __restrict__ xssm,
                            const float* __restrict__ proj, const float* __restrict__ A_log,
                            float* __restrict__ yscan)
{
    const int b = blockIdx.x >> 3;                       // D_INNER/256 = 8 blocks per batch
    const int d = ((blockIdx.x & 7) << 8) + threadIdx.x;

    float Ad[D_STATE], h[D_STATE];
    #pragma unroll
    for (int n = 0; n < D_STATE; ++n) {
        Ad[n] = -expf(A_log[d * D_STATE + n]);
        h[n]  = 0.0f;
    }

    const size_t rbase = (size_t)b * SEQ;
    for (int l = 0; l < SEQ; ++l) {
        const size_t row = rbase + l;
        const float dlt = delta[row * D_INNER + d];
        const float u   = xssm [row * D_INNER + d];

        const v4f* bp = (const v4f*)(proj + row * 96 + DT_RANK);             // Bm
        const v4f* cp = (const v4f*)(proj + row * 96 + DT_RANK + D_STATE);   // Cm
        v4f bvec[4], cvec[4];
        #pragma unroll
        for (int q = 0; q < 4; ++q) { bvec[q] = bp[q]; cvec[q] = cp[q]; }

        float y = 0.0f;
        #pragma unroll
        for (int n = 0; n < D_STATE; ++n) {
            const float Bn = bvec[n >> 2][n & 3];
            const float Cn = cvec[n >> 2][n & 3];
            const float dA = expf(dlt * Ad[n]);
            h[n] = dA * h[n] + dlt * Bn * u;
            y += h[n] * Cn;
        }
        yscan[row * D_INNER + d] = y;
    }
}

// ---------------------------------------------------------------------------
// y = (y_scan + x_ssm * D) * silu(z), in-place into y_scan.
// z = xz[:, D_INNER : 2*D_INNER].
// ---------------------------------------------------------------------------
__global__ void gate_kernel(float* __restrict__ yscan, const float* __restrict__ xssm,
                            const float* __restrict__ xz, const float* __restrict__ Dv)
{
    const int i = blockIdx.x * blockDim.x + threadIdx.x;
    if (i >= MROWS * D_INNER) return;
    const int d   = i & (D_INNER - 1);
    const int row = i >> 11;
    const float z = xz[(size_t)row * (2 * D_INNER) + D_INNER + d];
    const float y = yscan[i] + xssm[i] * Dv[d];
    yscan[i] = y * silu_f(z);
}

// ---------------------------------------------------------------------------
extern "C" void kernel_launch(void* const* d_in, const int* in_sizes, int n_in,
                              void* d_out, int out_size, void* d_ws, size_t ws_size,
                              hipStream_t stream)
{
    const float* x       = (const float*)d_in[0];
    const float* in_w    = (const float*)d_in[1];  // (2*D_INNER, D_MODEL)
    const float* conv_w  = (const float*)d_in[2];  // (D_INNER, 1, 4)
    const float* conv_b  = (const float*)d_in[3];
    const float* xproj_w = (const float*)d_in[4];  // (96, D_INNER)
    const float* dt_w    = (const float*)d_in[5];  // (D_INNER, DT_RANK)
    const float* dt_b    = (const float*)d_in[6];
    const float* A_log   = (const float*)d_in[7];  // (D_INNER, D_STATE)
    const float* Dv      = (const float*)d_in[8];
    const float* out_w   = (const float*)d_in[9];  // (D_MODEL, D_INNER)
    float* out = (float*)d_out;

    float* ws    = (float*)d_ws;
    float* xz    = ws;                                      // 4096 * 4096
    float* xssm  = xz    + (size_t)MROWS * 2 * D_INNER;     // 4096 * 2048
    float* delta = xssm  + (size_t)MROWS * D_INNER;         // 4096 * 2048
    float* ysc   = delta + (size_t)MROWS * D_INNER;         // 4096 * 2048
    float* proj  = ysc   + (size_t)MROWS * D_INNER;         // 4096 * 96

    const dim3 gblk(128);  // 4 wave32 -> 64x64 block patch (2x2 waves of 32x32)
    const int nelem = MROWS * D_INNER;

    // 1) xz = x @ in_proj_w.T          (M=4096, N=4096, K=1024)
    wmma_gemm_nt<<<dim3((2 * D_INNER) / 64, MROWS / 64), gblk, 0, stream>>>(
        x, in_w, xz, nullptr, MROWS, 2 * D_INNER, D_MODEL, D_MODEL, D_MODEL, 2 * D_INNER, 0);

    // 2) x_ssm = silu(causal_conv(x_in) + b)
    conv_silu_kernel<<<(nelem + 255) / 256, 256, 0, stream>>>(xz, conv_w, conv_b, xssm);

    // 3) proj = x_ssm @ x_proj_w.T     (M=4096, N=96, K=2048)
    wmma_gemm_nt<<<dim3((96 + 63) / 64, MROWS / 64), gblk, 0, stream>>>(
        xssm, xproj_w, proj, nullptr, MROWS, 96, D_INNER, D_INNER, D_INNER, 96, 0);

    // 4) delta = softplus(proj[:, :64] @ dt_proj_w.T + dt_b)   (M=4096, N=2048, K=64)
    wmma_gemm_nt<<<dim3(D_INNER / 64, MROWS / 64), gblk, 0, stream>>>(
        proj, dt_w, delta, dt_b, MROWS, D_INNER, DT_RANK, 96, DT_RANK, D_INNER, 1);

    // 5) selective scan
    scan_kernel<<<BATCH * (D_INNER / 256), 256, 0, stream>>>(delta, xssm, proj, A_log, ysc);

    // 6) gating (in-place into y_scan)
    gate_kernel<<<(nelem + 255) / 256, 256, 0, stream>>>(ysc, xssm, xz, Dv);

    // 7) out = y @ out_proj_w.T        (M=4096, N=1024, K=2048)
    wmma_gemm_nt<<<dim3(D_MODEL / 64, MROWS / 64), gblk, 0, stream>>>(
        ysc, out_w, out, nullptr, MROWS, D_MODEL, D_INNER, D_INNER, D_INNER, D_MODEL, 0);
}